// GFC_5763846111446
// MI455X (gfx1250) — compile-verified
//
#include <hip/hip_runtime.h>
#include <hip/hip_bf16.h>

// Problem constants (from reference setup_inputs)
#define BZ 32      // batch
#define SQ 32      // seq
#define DM 768     // hidden
#define NE 500000  // entities
#define NR 2000    // relations
#define NA 8192    // pairs per (t,b)
#define NT 2       // hops

typedef float v2f __attribute__((ext_vector_type(2)));
typedef float v8f __attribute__((ext_vector_type(8)));

__device__ __forceinline__ float wave_max32(float x) {
#pragma unroll
  for (int off = 16; off > 0; off >>= 1) x = fmaxf(x, __shfl_xor(x, off, 32));
  return x;
}
__device__ __forceinline__ float wave_sum32(float x) {
#pragma unroll
  for (int off = 16; off > 0; off >>= 1) x += __shfl_xor(x, off, 32);
  return x;
}
__device__ __forceinline__ float sigmoidf(float x) { return 1.0f / (1.0f + __expf(-x)); }
__device__ __forceinline__ float clamp1f(float x) { return x > 1.0f ? 1.0f : x; }

// ---------------------------------------------------------------- zero
__global__ void gfc_zero_k(float* __restrict__ p, int n) {
  for (int i = blockIdx.x * blockDim.x + threadIdx.x; i < n; i += gridDim.x * blockDim.x)
    p[i] = 0.0f;
}

// ---------------------------------------------------------------- big GEMM (WMMA f32 16x16x4)
// C[M x 768] = Amat[M x 768] @ W[768 x 768] + bias ; mode==1 -> sigmoid epilogue
// block = 256 threads (8 waves), block tile = 32(M) x 64(N), wave tile = 16x16
__global__ void gfc_gemm768_k(const float* __restrict__ Amat, const float* __restrict__ W,
                              const float* __restrict__ bias, float* __restrict__ C, int mode) {
  const int lane = threadIdx.x & 31;
  const int wave = threadIdx.x >> 5;
  const int wm = wave >> 2, wn = wave & 3;
  const int row0 = blockIdx.x * 32 + wm * 16;
  const int col0 = blockIdx.y * 64 + wn * 16;
  const int lr = lane & 15;
  const int hi = lane >> 4;  // K sub-pair select (K=0,1 vs K=2,3)

  const float* arow = Amat + (size_t)(row0 + lr) * DM;   // A frag: lane=M, x/y = K, K+1
  const float* bcol = W + col0 + lr;                     // B frag: lane=N, x/y = rows K, K+1
  v8f acc = {};
  for (int k = 0; k < DM; k += 4) {
    const int kk = k + 2 * hi;
    v2f a = {arow[kk], arow[kk + 1]};
    v2f b = {bcol[(size_t)kk * DM], bcol[(size_t)(kk + 1) * DM]};
    acc = __builtin_amdgcn_wmma_f32_16x16x4_f32(false, a, false, b, (short)0, acc, false, false);
  }
  const float bb = bias[col0 + lr];
#pragma unroll
  for (int v = 0; v < 8; ++v) {
    float val = acc[v] + bb;
    if (mode == 1) val = sigmoidf(val);
    C[(size_t)(row0 + hi * 8 + v) * DM + col0 + lr] = val;  // C frag: vgpr v -> M=v(+8), lane -> N
  }
}

// ---------------------------------------------------------------- q_logits (WMMA, B transposed)
// out[b,i,j] = dot(Q[b,i,:], Key[b,j,:]) ; per batch 32x32, K=768. block=128 (4 waves), grid=BZ
__global__ void gfc_qlogits_k(const float* __restrict__ Q, const float* __restrict__ Key,
                              float* __restrict__ out) {
  const int b = blockIdx.x;
  const int lane = threadIdx.x & 31;
  const int wave = threadIdx.x >> 5;  // 0..3
  const int row0 = (wave >> 1) * 16;
  const int col0 = (wave & 1) * 16;
  const int lr = lane & 15, hi = lane >> 4;
  const float* arow = Q + ((size_t)b * SQ + row0 + lr) * DM;
  const float* brow = Key + ((size_t)b * SQ + col0 + lr) * DM;  // B^T: lane=N picks Key row
  v8f acc = {};
  for (int k = 0; k < DM; k += 4) {
    const int kk = k + 2 * hi;
    v2f a = {arow[kk], arow[kk + 1]};
    v2f bv = {brow[kk], brow[kk + 1]};
    acc = __builtin_amdgcn_wmma_f32_16x16x4_f32(false, a, false, bv, (short)0, acc, false, false);
  }
#pragma unroll
  for (int v = 0; v < 8; ++v)
    out[(size_t)b * SQ * SQ + (size_t)(row0 + hi * 8 + v) * SQ + col0 + lr] = acc[v];
}

// ---------------------------------------------------------------- row softmax + mask + renorm (in place)
// one wave per (b,i) row of 32; grid = BZ*SQ/8 blocks of 256
__global__ void gfc_softmax_k(float* __restrict__ qd, const float* __restrict__ mask) {
  const int lane = threadIdx.x & 31;
  const int row = blockIdx.x * 8 + (threadIdx.x >> 5);  // b*SQ + i
  const int b = row >> 5;
  float x = qd[(size_t)row * SQ + lane];
  const float m = wave_max32(x);
  const float e = __expf(x - m);
  const float s = wave_sum32(e);
  float p = (e / s) * mask[b * SQ + lane];
  const float s2 = wave_sum32(p);
  qd[(size_t)row * SQ + lane] = p / (s2 + 1e-6f);
}

// ---------------------------------------------------------------- hop_ctx[b,i,d] = sum_j qd[b,i,j]*hop[b,j,d]
__global__ void gfc_hopctx_k(const float* __restrict__ qd, const float* __restrict__ hop,
                             float* __restrict__ ctx) {
  const int idx = blockIdx.x * blockDim.x + threadIdx.x;  // BZ*SQ*DM
  const int d = idx % DM;
  const int rest = idx / DM;
  const int i = rest & 31, b = rest >> 5;
  const float* qrow = qd + ((size_t)b * SQ + i) * SQ;
  const float* src = hop + (size_t)b * SQ * DM + d;
  float acc = 0.0f;
#pragma unroll
  for (int j = 0; j < SQ; ++j) acc += qrow[j] * src[(size_t)j * DM];
  ctx[idx] = acc;
}

// ---------------------------------------------------------------- hop-state updates
__global__ void gfc_upd0_k(const float* __restrict__ q, const float* __restrict__ ctx,
                           float* __restrict__ hop, float* __restrict__ prev) {
  const int i = blockIdx.x * blockDim.x + threadIdx.x;
  const float c = ctx[i];
  hop[i] = q[i] + c;
  prev[i] = c;
}
__global__ void gfc_upd1_k(const float* __restrict__ q, const float* __restrict__ ctx,
                           const float* __restrict__ z, float* __restrict__ prev,
                           float* __restrict__ hop) {
  const int i = blockIdx.x * blockDim.x + threadIdx.x;
  const float t = ctx[i] + z[i] * prev[i];
  hop[i] = q[i] + t;
  prev[i] = t;
}

// ---------------------------------------------------------------- word attention (one wave per batch)
__global__ void gfc_wordatt_k(const float* __restrict__ qd, const float* __restrict__ mask,
                              float* __restrict__ att) {
  const int lane = threadIdx.x & 31;
  const int b = blockIdx.x * 8 + (threadIdx.x >> 5);
  float c = 0.0f;
#pragma unroll
  for (int i = 0; i < SQ; ++i) c += qd[((size_t)b * SQ + i) * SQ + lane];
  const float m = wave_max32(c);
  const float e = __expf(c - m);
  const float s = wave_sum32(e);
  float p = (e / s) * mask[b * SQ + lane];
  const float s2 = wave_sum32(p);
  att[b * SQ + lane] = p / (s2 + 1e-6f);
}

// ---------------------------------------------------------------- ctx_h[b,d] + vw = ctx_h*w1 (fused)
__global__ void gfc_ctxh_k(const float* __restrict__ hop, const float* __restrict__ att,
                           const float* __restrict__ w1, float* __restrict__ ctx_h,
                           float* __restrict__ vw) {
  const int idx = blockIdx.x * blockDim.x + threadIdx.x;  // BZ*DM
  const int d = idx % DM, b = idx / DM;
  const float* arow = att + b * SQ;
  const float* src = hop + (size_t)b * SQ * DM + d;
  float acc = 0.0f;
#pragma unroll
  for (int j = 0; j < SQ; ++j) acc += arow[j] * src[(size_t)j * DM];
  ctx_h[idx] = acc;
  vw[idx] = acc * w1[d];
}

// ---------------------------------------------------------------- relation scores (WMMA, B transposed)
// proj[b,r] = dot(vw[b,:], rel_emb[r,:]) ; M=32 (2 wave tiles), N=2000 = 125*16, K=768
// block = 64 threads (2 waves), grid = 125
__global__ void gfc_proj_k(const float* __restrict__ vw, const float* __restrict__ relemb,
                           float* __restrict__ proj) {
  const int lane = threadIdx.x & 31;
  const int row0 = (threadIdx.x >> 5) * 16;
  const int col0 = blockIdx.x * 16;
  const int lr = lane & 15, hi = lane >> 4;
  const float* arow = vw + (size_t)(row0 + lr) * DM;
  const float* brow = relemb + (size_t)(col0 + lr) * DM;
  v8f acc = {};
  for (int k = 0; k < DM; k += 4) {
    const int kk = k + 2 * hi;
    v2f a = {arow[kk], arow[kk + 1]};
    v2f bv = {brow[kk], brow[kk + 1]};
    acc = __builtin_amdgcn_wmma_f32_16x16x4_f32(false, a, false, bv, (short)0, acc, false, false);
  }
#pragma unroll
  for (int v = 0; v < 8; ++v)
    proj[(size_t)(row0 + hi * 8 + v) * NR + col0 + lr] = acc[v];
}

// ---------------------------------------------------------------- per-pair gather/scatter
__global__ void gfc_scatter_k(const float* __restrict__ proj, const float* __restrict__ b1,
                              const float* __restrict__ pmask, const int* __restrict__ rels,
                              const int* __restrict__ psub, const int* __restrict__ pobj,
                              const float* __restrict__ src, int clamp_src,
                              float* __restrict__ dst) {
  const int idx = blockIdx.x * blockDim.x + threadIdx.x;  // BZ*NA
  const int b = idx >> 13;                                // / NA
  const float logit = proj[(size_t)b * NR + rels[idx]] + b1[0];
  const float p = sigmoidf(logit) * pmask[idx];
  float v = src[(size_t)b * NE + psub[idx]];
  if (clamp_src) v = clamp1f(v);  // last_e renorm of previous hop, applied on the fly
  atomicAdd(&dst[(size_t)b * NE + pobj[idx]], v * p);
}

// ---------------------------------------------------------------- hop attention over T=2
__global__ void gfc_hopattn_k(const float* __restrict__ ctx_h, const float* __restrict__ w_hop,
                              const float* __restrict__ b_hop, float* __restrict__ attn) {
  __shared__ float lg[BZ * NT];
  const int tid = threadIdx.x;  // 64
  const int b = tid >> 1, t = tid & 1;
  const float* row = ctx_h + ((size_t)t * BZ + b) * DM;
  float acc = 0.0f;
  for (int d = 0; d < DM; ++d) acc += row[d] * w_hop[d];
  lg[tid] = acc + b_hop[0];
  __syncthreads();
  if (t == 0) {
    const float l0 = lg[b * 2], l1 = lg[b * 2 + 1];
    const float m = fmaxf(l0, l1);
    const float e0 = __expf(l0 - m), e1 = __expf(l1 - m);
    const float s = e0 + e1;
    attn[b * 2] = e0 / s;
    attn[b * 2 + 1] = e1 / s;
  }
}

// ---------------------------------------------------------------- final clamp + blend (in place on out)
__global__ void gfc_combine_k(const float* __restrict__ e0, const float* __restrict__ attn,
                              float* __restrict__ out) {
  const size_t n = (size_t)BZ * NE;
  for (size_t i = (size_t)blockIdx.x * blockDim.x + threadIdx.x; i < n;
       i += (size_t)gridDim.x * blockDim.x) {
    const int b = (int)(i / NE);
    out[i] = attn[b * 2] * clamp1f(e0[i]) + attn[b * 2 + 1] * clamp1f(out[i]);
  }
}

extern "C" void kernel_launch(void* const* d_in, const int* in_sizes, int n_in,
                              void* d_out, int out_size, void* d_ws, size_t ws_size,
                              hipStream_t stream) {
  (void)in_sizes; (void)n_in; (void)out_size; (void)ws_size;
  const float* q_word_h  = (const float*)d_in[0];   // [B,S,D]
  const float* attn_mask = (const float*)d_in[1];   // [B,S]
  const float* heads     = (const float*)d_in[2];   // [B,E]
  const float* rel_emb   = (const float*)d_in[3];   // [R,D]
  const float* Wk        = (const float*)d_in[4];   // [D,D]
  const float* bk        = (const float*)d_in[5];   // [D]
  const float* w1        = (const float*)d_in[6];   // [D]
  const float* b1        = (const float*)d_in[7];   // [1]
  const float* Whw       = (const float*)d_in[8];   // [D,D]
  const float* bhw       = (const float*)d_in[9];   // [D]
  const float* w_hop     = (const float*)d_in[10];  // [D]
  const float* b_hop     = (const float*)d_in[11];  // [1]
  const float* pair_mask = (const float*)d_in[12];  // [T,B,A]
  const int*   pairs_sub = (const int*)d_in[13];    // [T,B,A]
  const int*   pairs_obj = (const int*)d_in[14];    // [T,B,A]
  const int*   rels      = (const int*)d_in[15];    // [T,B,A]
  float* out = (float*)d_out;                       // [B,E]

  float* ws = (float*)d_ws;
  size_t off = 0;
  float* hop_buf  = ws + off; off += (size_t)BZ * SQ * DM;  // q_word_h_hop
  float* prev_ctx = ws + off; off += (size_t)BZ * SQ * DM;
  float* h_key    = ws + off; off += (size_t)BZ * SQ * DM;
  float* hop_ctx  = ws + off; off += (size_t)BZ * SQ * DM;
  float* zgate    = ws + off; off += (size_t)BZ * SQ * DM;
  float* qdist    = ws + off; off += (size_t)BZ * SQ * SQ;
  float* q_att    = ws + off; off += (size_t)BZ * SQ;
  float* ctx_h    = ws + off; off += (size_t)NT * BZ * DM;  // [T][B][D]
  float* vw       = ws + off; off += (size_t)BZ * DM;
  float* proj     = ws + off; off += (size_t)BZ * NR;
  float* hop_attn = ws + off; off += (size_t)BZ * NT;
  float* e0       = ws + off; off += (size_t)BZ * NE;       // raw hop-0 scatter buffer

  // hop state starts as q_word_h; hop-1 result scatters straight into out (raw)
  hipMemcpyAsync(hop_buf, q_word_h, sizeof(float) * (size_t)BZ * SQ * DM,
                 hipMemcpyDeviceToDevice, stream);
  gfc_zero_k<<<4096, 256, 0, stream>>>(e0, BZ * NE);
  gfc_zero_k<<<4096, 256, 0, stream>>>(out, BZ * NE);

  for (int t = 0; t < NT; ++t) {
    // h_key = hop_buf @ Wk + bk  (M=1024, N=K=768)
    gfc_gemm768_k<<<dim3(BZ * SQ / 32, DM / 64), 256, 0, stream>>>(hop_buf, Wk, bk, h_key, 0);
    // q_logits / softmax / mask / renorm
    gfc_qlogits_k<<<BZ, 128, 0, stream>>>(q_word_h, h_key, qdist);
    gfc_softmax_k<<<BZ * SQ / 8, 256, 0, stream>>>(qdist, attn_mask);
    // hop_ctx from OLD hop state
    gfc_hopctx_k<<<(BZ * SQ * DM) / 256, 256, 0, stream>>>(qdist, hop_buf, hop_ctx);
    if (t == 0) {
      gfc_upd0_k<<<(BZ * SQ * DM) / 256, 256, 0, stream>>>(q_word_h, hop_ctx, hop_buf, prev_ctx);
    } else {
      gfc_gemm768_k<<<dim3(BZ * SQ / 32, DM / 64), 256, 0, stream>>>(prev_ctx, Whw, bhw, zgate, 1);
      gfc_upd1_k<<<(BZ * SQ * DM) / 256, 256, 0, stream>>>(q_word_h, hop_ctx, zgate, prev_ctx,
                                                           hop_buf);
    }
    // word attention -> ctx_h (on NEW hop state), fused vw = ctx_h * w1
    gfc_wordatt_k<<<BZ / 8, 256, 0, stream>>>(qdist, attn_mask, q_att);
    gfc_ctxh_k<<<(BZ * DM) / 256, 256, 0, stream>>>(hop_buf, q_att, w1,
                                                    ctx_h + (size_t)t * BZ * DM, vw);
    // dense relation score table (avoids 768 MB of embedding gathers)
    gfc_proj_k<<<NR / 16, 64, 0, stream>>>(vw, rel_emb, proj);
    // gather last_e at sub, sigmoid(proj)+mask weight, scatter-add at obj
    const float* src = (t == 0) ? heads : e0;
    float* dst = (t == 0) ? e0 : out;
    const size_t tb = (size_t)t * BZ * NA;
    gfc_scatter_k<<<(BZ * NA) / 256, 256, 0, stream>>>(proj, b1, pair_mask + tb, rels + tb,
                                                       pairs_sub + tb, pairs_obj + tb, src,
                                                       (t == 0) ? 0 : 1, dst);
  }

  gfc_hopattn_k<<<1, 64, 0, stream>>>(ctx_h, w_hop, b_hop, hop_attn);
  gfc_combine_k<<<4096, 256, 0, stream>>>(e0, hop_attn, out);
}